// ModelNew_80908593922669
// MI455X (gfx1250) — compile-verified
//
#include <hip/hip_runtime.h>

// ext-vector types for WMMA operands
typedef float v2f __attribute__((ext_vector_type(2)));
typedef float v8f __attribute__((ext_vector_type(8)));

#define B_TOT      128
#define C_OUT      24
#define NGRP       8
#define CPG        3          // channels per group
#define KTOT       81         // 3*3*3*3 reduction length
#define KPAD       84         // padded to multiple of 4
#define NS         12600      // 14*30*30 output positions per channel
#define TPG        105        // tile-groups of 8 per batch (840/8)
#define X_B_STRIDE 49152      // 3*16*32*32
#define EPSV       1e-5f

// compile-time patch offset for reduction index k = (ci,kd,kh,kw)
// strides: ci=16384 (16*32*32), kd=1024 (32*32), kh=32, kw=1
__host__ __device__ constexpr int patch_off(int k) {
    return (k >= KTOT) ? 0
         : (k / 27) * 16384 + ((k % 27) / 9) * 1024
         + (((k % 9) / 3)) * 32 + (k % 3);
}

// ---------------------------------------------------------------------------
// Kernel 0: zero the workspace accumulators (harness poisons ws with 0xAA)
// ---------------------------------------------------------------------------
__global__ void zero_ws_kernel(float* __restrict__ p, int n) {
    int i = blockIdx.x * blockDim.x + threadIdx.x;
    if (i < n) p[i] = 0.0f;
}

// ---------------------------------------------------------------------------
// Kernel 1: implicit-GEMM conv3d via V_WMMA_F32_16X16X4_F32, fused with
// per-channel sum / per-group sum-of-squares accumulation (y never stored).
// Block = 256 threads = 8 wave32; each wave does a 16-position x 16-channel
// D-tile (x2 N-tiles for 24 channels), K = 81 in 21 steps of 4.
// A-loads use constant instruction offsets off one per-lane base pointer;
// each lane loads the step's 4 K-values and cndmask-selects its half pair.
// ---------------------------------------------------------------------------
__global__ __launch_bounds__(256) void conv_gn_wmma_kernel(
    const float* __restrict__ x,      // [128,3,16,32,32]
    const float* __restrict__ wgt,    // [24,3,3,3,3] -> [24,81]
    const float* __restrict__ bias,   // [24]
    float* __restrict__ ws_sum,       // [128,24]
    float* __restrict__ ws_ssq)       // [128,8]
{
    // B-matrix staged [c][k], stride KPAD (=84 floats): k-pairs contiguous ->
    // ds_load_b64 pairs; 84-float stride banking is conflict-free.
    __shared__ float ldsw[32 * KPAD];   // 10.5 KB
    __shared__ float sAcc[C_OUT];       // block-level channel sums
    __shared__ float qAcc[NGRP];        // block-level group sum-of-squares

    const int tid = threadIdx.x;

    // ---- stage weights (zero-padded: c>=24 or k>=81 -> 0) ----
    for (int i = tid; i < 32 * KPAD; i += 256) {
        int c = i / KPAD, k = i % KPAD;
        float v = 0.0f;
        if (c < C_OUT && k < KTOT) v = wgt[c * KTOT + k];
        ldsw[i] = v;
    }
    if (tid < C_OUT) sAcc[tid] = 0.0f;
    if (tid < NGRP)  qAcc[tid] = 0.0f;
    __syncthreads();

    // ---- tile decode: all 8 waves of a block share one batch b ----
    const int b    = blockIdx.x / TPG;
    const int tg   = blockIdx.x % TPG;
    const int wave = tid >> 5;
    const int lane = tid & 31;
    const int tIdx = tg * 8 + wave;        // [0, 840)
    const int dp   = tIdx / 60;            // d' in [0,14)
    const int rem  = tIdx % 60;
    const int hp   = rem >> 1;             // h' in [0,30)
    const int half = rem & 1;              // w-tile half
    const int w0   = half << 4;            // w' base: 0 or 16

    const int  m   = lane & 15;            // A row / B column within tile
    const int  hi  = lane >> 4;            // K half (per ISA 16x4 f32 layout)
    const bool hib = (hi != 0);
    int wp = w0 + m; if (wp > 29) wp = 29; // clamp (garbage rows masked later)

    // per-lane base pointer; all k-offsets become constant IOFFSETs
    const float* __restrict__ xb =
        x + (b * X_B_STRIDE + dp * 1024 + hp * 32 + wp);

    // per-lane LDS base for B tiles (k-pair term 2*hi folded in once)
    const float* __restrict__ wb0 = &ldsw[m        * KPAD + 2 * hi];
    const float* __restrict__ wb1 = &ldsw[(m + 16) * KPAD + 2 * hi];

    const int c0 = m;                      // tile0 channel 0..15
    const int c1 = m + 16;                 // tile1 channel 16..31 (>=24 unused)
    const float bias0 = bias[c0];
    const float bias1 = bias[c1 < C_OUT ? c1 : (C_OUT - 1)];

    // fold conv bias into the accumulators: every row M gets +bias[ch]
    v8f acc0, acc1;
#pragma unroll
    for (int v = 0; v < 8; ++v) { acc0[v] = bias0; acc1[v] = bias1; }

    // ---- K loop: D = A(16x4 patches) x B(4x16 weights), 21 steps ----
#pragma unroll
    for (int t = 0; t < 21; ++t) {
        // constant-offset loads of this step's 4 K-values (same addr both halves)
        const float v0 = xb[patch_off(4 * t + 0)];
        const float v1 = xb[patch_off(4 * t + 1)];
        const float v2 = xb[patch_off(4 * t + 2)];
        const float v3 = xb[patch_off(4 * t + 3)];
        v2f a;
        a.x = hib ? v2 : v0;               // v_cndmask: lane half selects K pair
        a.y = hib ? v3 : v1;

        v2f b0 = *(const v2f*)&wb0[4 * t]; // ds_load_b64, imm offset 4t
        v2f b1 = *(const v2f*)&wb1[4 * t];

        acc0 = __builtin_amdgcn_wmma_f32_16x16x4_f32(
                   false, a, false, b0, (short)0, acc0, false, false);
        acc1 = __builtin_amdgcn_wmma_f32_16x16x4_f32(
                   false, a, false, b1, (short)0, acc1, false, false);
    }

    // ---- reduce D tiles: C/D layout lane l holds N=(l&15), rows M=v+8*hi ----
    float s0 = 0.f, q0 = 0.f, s1 = 0.f, q1 = 0.f;
#pragma unroll
    for (int v = 0; v < 8; ++v) {
        const int M = v + 8 * hi;
        if (w0 + M < 30) {                 // mask invalid / duplicated rows
            s0 += acc0[v]; q0 += acc0[v] * acc0[v];
            s1 += acc1[v]; q1 += acc1[v] * acc1[v];
        }
    }
    atomicAdd(&sAcc[c0], s0);
    atomicAdd(&qAcc[c0 / CPG], q0);
    if (c1 < C_OUT) {
        atomicAdd(&sAcc[c1], s1);
        atomicAdd(&qAcc[c1 / CPG], q1);
    }
    __syncthreads();

    // ---- one global atomic per channel/group per block ----
    if (tid < C_OUT) atomicAdd(&ws_sum[b * C_OUT + tid], sAcc[tid]);
    if (tid < NGRP)  atomicAdd(&ws_ssq[b * NGRP + tid], qAcc[tid]);
}

// ---------------------------------------------------------------------------
// Kernel 2: closed-form GroupNorm + global mean from (S, Q) moments.
// ---------------------------------------------------------------------------
__global__ void gn_finalize_kernel(
    const float* __restrict__ ws_sum, const float* __restrict__ ws_ssq,
    const float* __restrict__ gnw, const float* __restrict__ gnb,
    float* __restrict__ out)
{
    int b = blockIdx.x * blockDim.x + threadIdx.x;
    if (b >= B_TOT) return;
    const float ns  = (float)NS;
    const float cnt = (float)(CPG * NS);
    float acc = 0.0f;
#pragma unroll
    for (int g = 0; g < NGRP; ++g) {
        float S0 = ws_sum[b * C_OUT + 3 * g + 0];
        float S1 = ws_sum[b * C_OUT + 3 * g + 1];
        float S2 = ws_sum[b * C_OUT + 3 * g + 2];
        float mean = (S0 + S1 + S2) / cnt;
        float var  = ws_ssq[b * NGRP + g] / cnt - mean * mean;
        float inv  = rsqrtf(var + EPSV);
        acc += inv * (gnw[3 * g + 0] * (S0 - ns * mean) +
                      gnw[3 * g + 1] * (S1 - ns * mean) +
                      gnw[3 * g + 2] * (S2 - ns * mean));
    }
    float gb = 0.0f;
#pragma unroll
    for (int c = 0; c < C_OUT; ++c) gb += gnb[c];
    out[b] = acc / (24.0f * ns) + gb / 24.0f;
}

// ---------------------------------------------------------------------------
extern "C" void kernel_launch(void* const* d_in, const int* in_sizes, int n_in,
                              void* d_out, int out_size, void* d_ws, size_t ws_size,
                              hipStream_t stream) {
    const float* x   = (const float*)d_in[0];
    const float* cw  = (const float*)d_in[1];
    const float* cb  = (const float*)d_in[2];
    const float* gnw = (const float*)d_in[3];
    const float* gnb = (const float*)d_in[4];
    float* out = (float*)d_out;

    float* ws_sum = (float*)d_ws;                 // 128*24 floats
    float* ws_ssq = ws_sum + B_TOT * C_OUT;       // 128*8 floats
    const int nzero = B_TOT * C_OUT + B_TOT * NGRP;

    zero_ws_kernel<<<(nzero + 255) / 256, 256, 0, stream>>>(ws_sum, nzero);
    conv_gn_wmma_kernel<<<B_TOT * TPG, 256, 0, stream>>>(x, cw, cb, ws_sum, ws_ssq);
    gn_finalize_kernel<<<1, B_TOT, 0, stream>>>(ws_sum, ws_ssq, gnw, gnb, out);
}